// PLPContrastiveLoss_75797582840115
// MI455X (gfx1250) — compile-verified
//
#include <hip/hip_runtime.h>
#include <math.h>

#define B_  32
#define T_  8192
#define D_  256
#define NA  64          // MAX_ANCHORS
#define ROWPAD 260      // LDS row stride (floats), avoids bank conflicts

typedef float v2f __attribute__((ext_vector_type(2)));
typedef float v8f __attribute__((ext_vector_type(8)));

// ---------- online log-sum-exp helpers (branchless; EXEC stays full) ----------
__device__ __forceinline__ void lse_add(float& m, float& s, float v) {
    // add exp(v); v == -1e35f acts as "masked out" (underflows to 0)
    float nm = fmaxf(m, v);
    s = s * __expf(m - nm) + __expf(v - nm);
    m = nm;
}
__device__ __forceinline__ void lse_merge(float& m, float& s, float om, float os) {
    float nm = fmaxf(m, om);
    s = s * __expf(m - nm) + os * __expf(om - nm);
    m = nm;
}
__device__ __forceinline__ float lse_final(float m, float s) {
    float mm = (m <= -5e29f) ? 0.0f : m;            // reference NEG/2 rule
    return mm + __logf(fmaxf(s, 1e-30f));
}

// ---------- K1: stable stream-compaction of peaks, per batch ----------
__global__ void k_compact(const float* __restrict__ pm, const float* __restrict__ pv,
                          const float* __restrict__ vm,
                          int* __restrict__ frames, float* __restrict__ w,
                          int* __restrict__ count) {
    int b = blockIdx.x, tid = threadIdx.x;          // 256 threads
    __shared__ int scan[256];
    __shared__ int base_s;
    if (tid == 0) base_s = 0;
    __syncthreads();
    for (int c0 = 0; c0 < T_; c0 += 256) {
        int t = c0 + tid;
        int p = (pm[b * T_ + t] > 0.f && vm[b * T_ + t] > 0.f) ? 1 : 0;
        scan[tid] = p;
        __syncthreads();
        for (int off = 1; off < 256; off <<= 1) {
            int v = (tid >= off) ? scan[tid - off] : 0;
            __syncthreads();
            scan[tid] += v;
            __syncthreads();
        }
        int incl = scan[tid];
        int base = base_s;
        if (p) {
            int idx = base + incl - 1;
            frames[b * T_ + idx] = t;
            w[b * T_ + idx] = fmaxf(pv[b * T_ + t], 1e-6f);
        }
        __syncthreads();
        if (tid == 255) base_s = base + incl;
        __syncthreads();
    }
    int cnt = base_s;
    if (tid == 0) count[b] = cnt;
    for (int s = cnt + tid; s < T_; s += 256) { frames[b * T_ + s] = 0; w[b * T_ + s] = 0.f; }
}

// ---------- K2: inverse L2 norm of gathered embedding rows ----------
__global__ void k_invnorm(const float* __restrict__ emb, const int* __restrict__ frames,
                          const int* __restrict__ count, float* __restrict__ invn) {
    int b = blockIdx.y;
    int s = blockIdx.x * 256 + threadIdx.x;
    if (s >= T_) return;
    int cnt = count[b];
    if (s < cnt) {
        int f = frames[b * T_ + s];
        const float* row = emb + ((size_t)b * T_ + f) * D_;
        float acc = 0.f;
        for (int k = 0; k < D_; k += 4) {
            float4 v = *(const float4*)(row + k);
            acc += v.x * v.x + v.y * v.y + v.z * v.z + v.w * v.w;
        }
        invn[b * T_ + s] = 1.0f / fmaxf(sqrtf(acc), 1e-12f);
    } else {
        invn[b * T_ + s] = 0.f;
    }
}

// ---------- K3: top-64 anchors by weight + per-anchor tau (median interval) ----------
__global__ void k_anchors(const float* __restrict__ w, const int* __restrict__ frames,
                          const int* __restrict__ count,
                          int* __restrict__ apslot, float* __restrict__ tau_o,
                          float* __restrict__ aframe, float* __restrict__ wa,
                          int* __restrict__ aok) {
    int b = blockIdx.x, tid = threadIdx.x;          // 256 threads
    __shared__ unsigned taken[T_ / 32];
    __shared__ float rv[256];
    __shared__ int   ri[256];
    int cnt = count[b];
    for (int i = tid; i < T_ / 32; i += 256) taken[i] = 0u;
    __syncthreads();
    for (int a = 0; a < NA; a++) {
        float bv = -3e30f; int bi = T_;
        for (int s = tid; s < T_; s += 256) {
            if (taken[s >> 5] & (1u << (s & 31))) continue;
            float v = (s < cnt) ? w[b * T_ + s] : -1e30f;   // -inf sentinel for invalid
            if (v > bv) { bv = v; bi = s; }                 // ascending scan: lowest idx on ties
        }
        rv[tid] = bv; ri[tid] = bi;
        __syncthreads();
        for (int off = 128; off > 0; off >>= 1) {
            if (tid < off) {
                float ov = rv[tid + off]; int oi = ri[tid + off];
                if (ov > rv[tid] || (ov == rv[tid] && oi < ri[tid])) { rv[tid] = ov; ri[tid] = oi; }
            }
            __syncthreads();
        }
        if (tid == 0) {
            int bi0 = ri[0];
            apslot[b * NA + a] = bi0;
            taken[bi0 >> 5] |= (1u << (bi0 & 31));
        }
        __syncthreads();
    }
    if (tid < NA) {
        int a = tid;
        int pa = apslot[b * NA + a];
        bool avalid = pa < cnt;
        int start = max(pa - 2, 0);
        int endv  = min(cnt - 1, pa + 3);
        float vals[5]; int nval = 0;
        for (int j = 0; j < 5; j++) {
            int widx = start + j;
            bool okj = widx < endv;
            int c = min(widx, T_ - 2);
            float iv = (float)(frames[b * T_ + c + 1] - frames[b * T_ + c]);
            if (okj && iv > 0.f) vals[nval++] = iv;
        }
        for (int i = 1; i < nval; i++) {                     // tiny insertion sort
            float key = vals[i]; int j = i - 1;
            while (j >= 0 && vals[j] > key) { vals[j + 1] = vals[j]; j--; }
            vals[j + 1] = key;
        }
        bool tok = nval > 0;
        float tau = tok ? vals[max((nval - 1) / 2, 0)] : 1.0f;
        tau_o[b * NA + a]  = tau;
        aframe[b * NA + a] = (float)frames[b * T_ + pa];
        wa[b * NA + a]     = w[b * T_ + pa];
        aok[b * NA + a]    = (avalid && tok) ? 1 : 0;
    }
}

// ---------- K4: fused f32 WMMA GEMM (anchor x slot sims) + online masked LSE ----------
__global__ __launch_bounds__(32)
void k_gemm_lse(const float* __restrict__ emb, const int* __restrict__ frames,
                const float* __restrict__ invn, const int* __restrict__ count,
                const int* __restrict__ apslot, const float* __restrict__ tau_v,
                const float* __restrict__ aframe_v, const float* __restrict__ wa_v,
                const int* __restrict__ aok_v,
                float* __restrict__ pairwl, float* __restrict__ pairw) {
    int bid = blockIdx.x;
    int b = bid >> 2;
    int atile = bid & 3;                 // 4 tiles of 16 anchors
    int lane = threadIdx.x;
    int n = lane & 15, hi = lane >> 4;
    __shared__ float lds_a[16 * ROWPAD];
    __shared__ float lds_b[16 * ROWPAD];
    int cnt = count[b];

    // Stage normalized A tile (16 anchor rows x 256) into LDS
    for (int i = lane; i < 16 * (D_ / 4); i += 32) {
        int row = i >> 6, c4 = (i & 63) * 4;
        int a = atile * 16 + row;
        int slot = apslot[b * NA + a];
        float sc = invn[b * T_ + slot];          // 0 for invalid anchors -> zero row
        int f = frames[b * T_ + slot];
        float4 v = *(const float4*)(emb + ((size_t)b * T_ + f) * D_ + c4);
        float* dst = &lds_a[row * ROWPAD + c4];
        dst[0] = v.x * sc; dst[1] = v.y * sc; dst[2] = v.z * sc; dst[3] = v.w * sc;
    }

    // Per-lane anchor params for the 8 anchors this lane accumulates
    float p_tau[8], p_frame[8]; int p_slot[8];
    for (int r = 0; r < 8; r++) {
        int a = atile * 16 + r + 8 * hi;
        p_tau[r]   = tau_v[b * NA + a];
        p_frame[r] = aframe_v[b * NA + a];
        p_slot[r]  = apslot[b * NA + a];
    }
    float mP[8], sP[8], mN[8], sN[8], mB[8], sB[8];
    int cP[8], cN[8], cB[8];
    for (int r = 0; r < 8; r++) {
        mP[r] = mN[r] = mB[r] = -1e30f;
        sP[r] = sN[r] = sB[r] = 0.f;
        cP[r] = cN[r] = cB[r] = 0;
    }

    int nqt = (cnt + 15) >> 4;
    for (int qt = 0; qt < nqt; qt++) {
        __syncthreads();
        // Stage normalized B tile (16 slot rows x 256) into LDS
        for (int i = lane; i < 16 * (D_ / 4); i += 32) {
            int row = i >> 6, c4 = (i & 63) * 4;
            int slot = min(qt * 16 + row, cnt - 1);
            float sc = invn[b * T_ + slot];
            int f = frames[b * T_ + slot];
            float4 v = *(const float4*)(emb + ((size_t)b * T_ + f) * D_ + c4);
            float* dst = &lds_b[row * ROWPAD + c4];
            dst[0] = v.x * sc; dst[1] = v.y * sc; dst[2] = v.z * sc; dst[3] = v.w * sc;
        }
        __syncthreads();

        // 16x16 sims tile, K=256 chained as 64 x V_WMMA_F32_16X16X4_F32
        v8f acc = {};
        const float* pa_ = &lds_a[n * ROWPAD + 2 * hi];
        const float* pb_ = &lds_b[n * ROWPAD + 2 * hi];
        #pragma unroll 8
        for (int k = 0; k < D_; k += 4) {
            v2f af = *(const v2f*)(pa_ + k);
            v2f bf = *(const v2f*)(pb_ + k);
            acc = __builtin_amdgcn_wmma_f32_16x16x4_f32(
                false, af, false, bf, (short)0, acc, false, false);
        }

        // Online masked LSE update (branchless selects only)
        int q = qt * 16 + n;
        bool qv = q < cnt;
        int qc = min(q, cnt - 1);
        float fq = (float)frames[b * T_ + qc];
        #pragma unroll
        for (int r = 0; r < 8; r++) {
            float sim  = acc[r] * 10.0f;             // / TEMPERATURE
            float tau  = p_tau[r];
            float dist = fabsf(fq - p_frame[r]);
            bool base  = qv && (q != p_slot[r]);
            float t1 = tau,        tol1 = fmaxf(2.0f, t1 * 0.25f);
            float t2 = 2.f * tau,  tol2 = fmaxf(2.0f, t2 * 0.25f);
            bool pos = ((fabsf(dist - t1) <= tol1) || (fabsf(dist - t2) <= tol2)) && base;
            float radius = fmaxf(4.0f, tau * 0.5f);
            bool sneg = (dist > radius) && base && !pos;
            lse_add(mP[r], sP[r], pos  ? sim : -1e35f);
            lse_add(mN[r], sN[r], sneg ? sim : -1e35f);
            lse_add(mB[r], sB[r], base ? sim : -1e35f);
            cP[r] += pos ? 1 : 0; cN[r] += sneg ? 1 : 0; cB[r] += base ? 1 : 0;
        }
    }

    // Cross-lane LSE reduction over the 16 q-lanes of each half-wave
    for (int off = 1; off < 16; off <<= 1) {
        #pragma unroll
        for (int r = 0; r < 8; r++) {
            float om, os;
            om = __shfl_xor(mP[r], off, 32); os = __shfl_xor(sP[r], off, 32);
            lse_merge(mP[r], sP[r], om, os);
            om = __shfl_xor(mN[r], off, 32); os = __shfl_xor(sN[r], off, 32);
            lse_merge(mN[r], sN[r], om, os);
            om = __shfl_xor(mB[r], off, 32); os = __shfl_xor(sB[r], off, 32);
            lse_merge(mB[r], sB[r], om, os);
            cP[r] += __shfl_xor(cP[r], off, 32);
            cN[r] += __shfl_xor(cN[r], off, 32);
            cB[r] += __shfl_xor(cB[r], off, 32);
        }
    }

    if (n == 0) {   // lanes 0 and 16 hold anchors r+0 / r+8
        for (int r = 0; r < 8; r++) {
            int a = atile * 16 + r + 8 * hi;
            float lse_pos = lse_final(mP[r], sP[r]);
            float md = mP[r], sd = sP[r];
            if (cN[r] > 0) lse_merge(md, sd, mN[r], sN[r]);   // pos U strict-neg
            else         { md = mB[r]; sd = sB[r]; }          // fallback: all base_ok
            float lse_den = lse_final(md, sd);
            bool posany = cP[r] > 0;
            bool negany = (cN[r] > 0) || (cB[r] - cP[r] > 0);
            bool ok = (aok_v[b * NA + a] != 0) && posany && negany;
            float wv = wa_v[b * NA + a];
            float pl = lse_den - lse_pos;
            pairwl[b * NA + a] = ok ? pl * wv : 0.f;
            pairw[b * NA + a]  = ok ? wv      : 0.f;
        }
    }
}

// ---------- K5: deterministic final reduction ----------
__global__ void k_reduce(const float* __restrict__ pairwl, const float* __restrict__ pairw,
                         float* __restrict__ out) {
    __shared__ float swl[256], sw[256];
    int tid = threadIdx.x;
    float awl = 0.f, aw = 0.f;
    for (int i = tid; i < B_ * NA; i += 256) { awl += pairwl[i]; aw += pairw[i]; }
    swl[tid] = awl; sw[tid] = aw;
    __syncthreads();
    for (int off = 128; off > 0; off >>= 1) {
        if (tid < off) { swl[tid] += swl[tid + off]; sw[tid] += sw[tid + off]; }
        __syncthreads();
    }
    if (tid == 0) out[0] = (sw[0] > 0.f) ? swl[0] / fmaxf(sw[0], 1e-30f) : 0.f;
}

extern "C" void kernel_launch(void* const* d_in, const int* in_sizes, int n_in,
                              void* d_out, int out_size, void* d_ws, size_t ws_size,
                              hipStream_t stream) {
    const float* emb = (const float*)d_in[0];
    const float* pm  = (const float*)d_in[1];
    const float* pv  = (const float*)d_in[2];
    const float* vm  = (const float*)d_in[3];
    float* out = (float*)d_out;

    char* p = (char*)d_ws;
    int*   d_count  = (int*)p;   p += 256;
    int*   d_frames = (int*)p;   p += (size_t)B_ * T_ * 4;
    float* d_w      = (float*)p; p += (size_t)B_ * T_ * 4;
    float* d_invn   = (float*)p; p += (size_t)B_ * T_ * 4;
    int*   d_apslot = (int*)p;   p += B_ * NA * 4;
    float* d_tau    = (float*)p; p += B_ * NA * 4;
    float* d_aframe = (float*)p; p += B_ * NA * 4;
    float* d_wa     = (float*)p; p += B_ * NA * 4;
    int*   d_aok    = (int*)p;   p += B_ * NA * 4;
    float* d_pairwl = (float*)p; p += B_ * NA * 4;
    float* d_pairw  = (float*)p; p += B_ * NA * 4;

    k_compact<<<B_, 256, 0, stream>>>(pm, pv, vm, d_frames, d_w, d_count);
    k_invnorm<<<dim3(T_ / 256, B_), 256, 0, stream>>>(emb, d_frames, d_count, d_invn);
    k_anchors<<<B_, 256, 0, stream>>>(d_w, d_frames, d_count,
                                      d_apslot, d_tau, d_aframe, d_wa, d_aok);
    k_gemm_lse<<<B_ * 4, 32, 0, stream>>>(emb, d_frames, d_invn, d_count,
                                          d_apslot, d_tau, d_aframe, d_wa, d_aok,
                                          d_pairwl, d_pairw);
    k_reduce<<<1, 256, 0, stream>>>(d_pairwl, d_pairw, out);
}